// MultiHeadAttentionQuantum_65481071405162
// MI455X (gfx1250) — compile-verified
//
#include <hip/hip_runtime.h>
#include <math.h>

typedef __attribute__((ext_vector_type(2))) float v2f;
typedef __attribute__((ext_vector_type(8))) float v8f;

#define B_  16
#define S_  1024
#define E_  8
#define NH  4
#define QSPLIT 8   // query-tile groups per (b,head) -> 512 blocks total

// ---------------------------------------------------------------------------
// Kernel 1: closed-form quantum head.
// out[w] = prod_{i<=w} cos(x_i+theta_i) (w>=1); out[0] = prod_{i=1..7}.
// Writes h in [B][NH][S][2] float layout (per-(b,head) contiguous 8KB tiles).
// ---------------------------------------------------------------------------
__global__ __launch_bounds__(256)
void qhead_kernel(const float* __restrict__ x, const float* __restrict__ theta,
                  float* __restrict__ h) {
  int n = blockIdx.x * blockDim.x + threadIdx.x;
  if (n >= B_ * S_) return;
  const float4* xr = (const float4*)(x + n * 8);
  float4 xa = xr[0], xb = xr[1];
  float c[8];
  c[0] = __cosf(xa.x + theta[0]); c[1] = __cosf(xa.y + theta[1]);
  c[2] = __cosf(xa.z + theta[2]); c[3] = __cosf(xa.w + theta[3]);
  c[4] = __cosf(xb.x + theta[4]); c[5] = __cosf(xb.y + theta[5]);
  c[6] = __cosf(xb.z + theta[6]); c[7] = __cosf(xb.w + theta[7]);
  float out[8];
  float p = c[0];
#pragma unroll
  for (int w = 1; w < 8; ++w) { p *= c[w]; out[w] = p; }
  out[0] = c[1]*c[2]*c[3]*c[4]*c[5]*c[6]*c[7];
  int b = n >> 10, s = n & 1023;
  float2* hp = (float2*)h;
#pragma unroll
  for (int hd = 0; hd < NH; ++hd)
    hp[(b*NH + hd)*S_ + s] = make_float2(out[2*hd], out[2*hd + 1]);
}

// ---------------------------------------------------------------------------
// Kernel 2: fused attention per (b,head), h tile in LDS.
// Scores via V_WMMA_F32_16X16X4_F32 (d=2 zero-padded to K=4).
// |h|<=1 => scores in [-sqrt2, sqrt2]: softmax without max-shift is exact.
// Wave owns one 16-query tile; accumulators stay in WMMA C/D lane layout,
// 16-lane column reduction deferred to the end.
// ---------------------------------------------------------------------------
__global__ __launch_bounds__(256)
void attn_kernel(const float* __restrict__ h, const int* __restrict__ mask,
                 float* __restrict__ O) {
  __shared__ float2 hsh[S_];
  __shared__ float  msh[S_];

  int blk  = blockIdx.x;          // B*NH*QSPLIT
  int qs   = blk % QSPLIT;
  int bh   = blk / QSPLIT;
  int b    = bh >> 2;
  int head = bh & 3;

  const float2* hp = ((const float2*)h) + bh * S_;
  const int*    mp = mask + b * S_;
  for (int i = threadIdx.x; i < S_; i += 256) {
    hsh[i] = hp[i];
    msh[i] = (mp[i] == 0) ? 0.0f : 1.0f;
  }
  __syncthreads();

  int lane = threadIdx.x & 31;
  int wid  = threadIdx.x >> 5;    // 0..7
  int l    = lane & 15;           // N index / row-in-tile
  int hi   = lane >> 4;           // half-wave select
  int q0   = (qs * 8 + wid) * 16;

  const float scale = 0.70710678118654752f;   // 1/sqrt(HEAD_DIM)
  // A-matrix 16x4: lanes 0-15 hold K=0,1 (VGPR0,1); lanes 16-31 hold K=2,3 = 0.
  float2 qv = hsh[q0 + l];
  v2f A;
  A[0] = hi ? 0.0f : qv.x * scale;
  A[1] = hi ? 0.0f : qv.y * scale;

  float accd[8], accx[8], accy[8];
#pragma unroll
  for (int r = 0; r < 8; ++r) { accd[r] = 0.f; accx[r] = 0.f; accy[r] = 0.f; }

  for (int kt = 0; kt < 64; ++kt) {
    int k0 = kt * 16;
    float2 hk = hsh[k0 + l];      // key for column N=l (both halves)
    float  mk = msh[k0 + l];
    // B-matrix 4x16: lanes 0-15 hold rows K=0,1; lanes 16-31 rows K=2,3 = 0.
    v2f Bv;
    Bv[0] = hi ? 0.0f : hk.x;
    Bv[1] = hi ? 0.0f : hk.y;
    v8f Cz = {0.f, 0.f, 0.f, 0.f, 0.f, 0.f, 0.f, 0.f};
    v8f sD = __builtin_amdgcn_wmma_f32_16x16x4_f32(
        false, A, false, Bv, (short)0, Cz, false, false);
#pragma unroll
    for (int r = 0; r < 8; ++r) {
      float e = __expf(sD[r]) * mk;     // mask==0 -> weight 0 (== -inf score)
      accd[r] += e;
      accx[r] += e * hk.x;
      accy[r] += e * hk.y;
    }
  }

  // Column (N) reduction across the 16 lanes of each half-wave.
#pragma unroll
  for (int r = 0; r < 8; ++r) {
    float d = accd[r], xx = accx[r], yy = accy[r];
#pragma unroll
    for (int off = 8; off >= 1; off >>= 1) {
      d  += __shfl_xor(d,  off, 32);
      xx += __shfl_xor(xx, off, 32);
      yy += __shfl_xor(yy, off, 32);
    }
    if (l == 0) {
      int q = q0 + r + 8 * hi;
      float inv = 1.0f / d;
      ((float2*)O)[(b * S_ + q) * NH + head] = make_float2(xx * inv, yy * inv);
    }
  }
}

// ---------------------------------------------------------------------------
// Kernel 3: y = O @ w_out^T   (8x8, w_out row-major: y_j = sum_i O_i * w[j*8+i])
// ---------------------------------------------------------------------------
__global__ __launch_bounds__(256)
void proj_kernel(const float* __restrict__ O, const float* __restrict__ w,
                 float* __restrict__ y) {
  __shared__ float wsh[64];
  if (threadIdx.x < 64) wsh[threadIdx.x] = w[threadIdx.x];
  __syncthreads();
  int n = blockIdx.x * blockDim.x + threadIdx.x;
  if (n >= B_ * S_) return;
  float4 a = ((const float4*)O)[n * 2];
  float4 bq = ((const float4*)O)[n * 2 + 1];
  float o[8] = {a.x, a.y, a.z, a.w, bq.x, bq.y, bq.z, bq.w};
  float r[8];
#pragma unroll
  for (int j = 0; j < 8; ++j) {
    float acc = 0.f;
#pragma unroll
    for (int i = 0; i < 8; ++i) acc += o[i] * wsh[j * 8 + i];
    r[j] = acc;
  }
  float4 r0 = {r[0], r[1], r[2], r[3]};
  float4 r1 = {r[4], r[5], r[6], r[7]};
  ((float4*)y)[n * 2]     = r0;
  ((float4*)y)[n * 2 + 1] = r1;
}

// ---------------------------------------------------------------------------
extern "C" void kernel_launch(void* const* d_in, const int* in_sizes, int n_in,
                              void* d_out, int out_size, void* d_ws, size_t ws_size,
                              hipStream_t stream) {
  const float* x     = (const float*)d_in[0];   // (16,1024,8) f32
  const int*   mask  = (const int*)  d_in[1];   // (16,1024)   i32
  const float* theta = (const float*)d_in[2];   // (8,)        f32
  const float* w_out = (const float*)d_in[3];   // (8,8)       f32
  float*       yout  = (float*)d_out;           // (16,1024,8) f32

  float* h = (float*)d_ws;                              // 512 KB: [B][NH][S][2]
  float* O = (float*)((char*)d_ws + (size_t)B_*NH*S_*2*sizeof(float)); // 512 KB: [B][S][8]

  int nrows = B_ * S_;
  qhead_kernel<<<(nrows + 255) / 256, 256, 0, stream>>>(x, theta, h);
  attn_kernel<<<B_ * NH * QSPLIT, 256, 0, stream>>>(h, mask, O);
  proj_kernel<<<(nrows + 255) / 256, 256, 0, stream>>>(O, w_out, yout);
}